// model_weight_70566312673847
// MI455X (gfx1250) — compile-verified
//
#include <hip/hip_runtime.h>

// ---------------------------------------------------------------------------
// Problem constants (x: [32, 1, 16, 288, 384] f32)
// ---------------------------------------------------------------------------
namespace {
constexpr int   kB   = 32;
constexpr int   kT   = 16;
constexpr int   kHW  = 288 * 384;   // 110592 pixels per (t,b) map
constexpr int   kHW4 = kHW / 4;     // 27648 float4 groups
constexpr float kN   = (float)kHW;

// Workspace layout (float offsets)
constexpr int OFF_SUM_S = 0;      // [512] f32   per-(t,b) sum of s
constexpr int OFF_SQ_S  = 512;    // [512] f32   per-(t,b) sum of s^2
constexpr int OFF_MIN_S = 1024;   // [512] u32   ordered-key min
constexpr int OFF_MAX_S = 1536;   // [512] u32   ordered-key max
constexpr int OFF_SUM_G = 2048;   // [32]  f32   per-b sum of gt
constexpr int OFF_SQ_G  = 2080;   // [32]  f32
constexpr int OFF_MIN_G = 2112;   // [32]  u32
constexpr int OFF_MAX_G = 2144;   // [32]  u32
constexpr int OFF_PROD  = 2176;   // [512] f32   per-(t,b) sum of s*gt
constexpr int OFF_SIM   = 2688;   // [512] f32   per-(t,b) sim accumulator
constexpr int OFF_GT    = 4096;   // [kB*kHW] f32 gt map (T-mean), optional
constexpr int INIT_WORDS = 3200;
}

typedef float f4  __attribute__((ext_vector_type(4)));
typedef float v2f __attribute__((ext_vector_type(2)));
typedef float v8f __attribute__((ext_vector_type(8)));

// Order-preserving float <-> uint key (for atomicMin/Max over signed floats)
__device__ __forceinline__ unsigned fkey(float f) {
  unsigned u = __float_as_uint(f);
  return (u & 0x80000000u) ? ~u : (u | 0x80000000u);
}
__device__ __forceinline__ float funkey(unsigned k) {
  unsigned u = (k & 0x80000000u) ? (k & 0x7FFFFFFFu) : ~k;
  return __uint_as_float(u);
}

__device__ __forceinline__ f4 ntload(const f4* __restrict__ p) {
  return __builtin_nontemporal_load(p);   // NT: keep gt resident in L2
}

// ---------------------------------------------------------------------------
// Wave32 full-lane sum via V_WMMA_F32_16X16X4_F32 with ones-B.
// A layout (ISA 7.12.2): lane L<16 -> A[L][0],A[L][1] ; lane L>=16 -> A[L-16][2..3]
// With a.y=0 and B=1s:  D[m][n] = v[m] + v[m+16]  (for every n)
// D layout: vgpr r, lane L<16 -> D[r][L]; lane L>=16 -> D[r+8][L-16]
// => per-lane sum of 8 D vgprs + swap-halves add == full 32-lane sum.
// Ones-B is permutation invariant, so exact B lane layout is irrelevant.
// ---------------------------------------------------------------------------
__device__ __forceinline__ float wave_sum_wmma(float v) {
  v2f a;    a[0] = v;    a[1] = 0.0f;
  v2f ones; ones[0] = 1.0f; ones[1] = 1.0f;
  v8f c = {0.f, 0.f, 0.f, 0.f, 0.f, 0.f, 0.f, 0.f};
  v8f d = __builtin_amdgcn_wmma_f32_16x16x4_f32(
      false, a, false, ones, (short)0, c, false, false);
  float r = ((d[0] + d[1]) + (d[2] + d[3])) + ((d[4] + d[5]) + (d[6] + d[7]));
  r += __shfl_xor(r, 16, 32);
  return r;   // every lane holds the full sum
}

__device__ __forceinline__ float wave_min(float v) {
#pragma unroll
  for (int o = 16; o > 0; o >>= 1) v = fminf(v, __shfl_xor(v, o, 32));
  return v;
}
__device__ __forceinline__ float wave_max(float v) {
#pragma unroll
  for (int o = 16; o > 0; o >>= 1) v = fmaxf(v, __shfl_xor(v, o, 32));
  return v;
}

// gt loader: from ws map, or recompute the T-mean on the fly (ws too small)
template <bool GT_WS>
__device__ __forceinline__ f4 load_gt(const f4* __restrict__ x4,
                                      const f4* __restrict__ gt4,
                                      int b, int p4) {
  if (GT_WS) {
    return gt4[(size_t)b * kHW4 + p4];
  } else {
    const f4* xp = x4 + (size_t)b * (kT * kHW4) + p4;
    f4 acc = (f4)0.0f;
#pragma unroll
    for (int t = 0; t < kT; ++t) acc += ntload(xp + (size_t)t * kHW4);
    return acc * 0.0625f;
  }
}

// ---------------------------------------------------------------------------
// K0: initialize accumulators (graph-replay safe: runs every launch)
// ---------------------------------------------------------------------------
__global__ __launch_bounds__(256) void k_init(unsigned* __restrict__ ws) {
  int i = blockIdx.x * 256 + threadIdx.x;
  if (i < INIT_WORDS) {
    bool isMin = (i >= OFF_MIN_S && i < OFF_MAX_S) ||
                 (i >= OFF_MIN_G && i < OFF_MIN_G + 32);
    ws[i] = isMin ? 0xFFFFFFFFu : 0u;   // 0u == 0.0f for sums, and min-key for max
  }
}

// ---------------------------------------------------------------------------
// K1: one streaming pass over x. Per block: batch b, 4096-pixel chunk.
//     Produces gt map (optional) + per-(t,b) {sum, sumsq, min, max}.
// ---------------------------------------------------------------------------
template <bool GT_WS>
__global__ __launch_bounds__(256) void k_pass_a(const float* __restrict__ xf,
                                                float* __restrict__ ws) {
  const f4* x4   = (const f4*)xf;
  float*    sumS = ws + OFF_SUM_S;
  float*    sqS  = ws + OFF_SQ_S;
  unsigned* minS = (unsigned*)(ws + OFF_MIN_S);
  unsigned* maxS = (unsigned*)(ws + OFF_MAX_S);
  f4*       gt4  = (f4*)(ws + OFF_GT);

  const int tid = threadIdx.x, lane = tid & 31, wv = tid >> 5;
  const int chunk = blockIdx.x;           // 0..26
  const int b     = blockIdx.y;           // 0..31
  const int base4 = chunk * 1024;

  __shared__ float st[kT][8][4];

  f4 gacc[4];
#pragma unroll
  for (int i = 0; i < 4; ++i) gacc[i] = (f4)0.0f;

#pragma unroll 1
  for (int t = 0; t < kT; ++t) {
    const f4* p = x4 + (size_t)(b * kT + t) * kHW4 + base4 + tid;
    float ls = 0.f, lsq = 0.f, lmin = 3.402823e38f, lmax = -3.402823e38f;
#pragma unroll
    for (int i = 0; i < 4; ++i) {
      f4 v = ntload(p + i * 256);
      gacc[i] += v;
      ls  += (v.x + v.y) + (v.z + v.w);
      lsq += (v.x * v.x + v.y * v.y) + (v.z * v.z + v.w * v.w);
      lmin = fminf(lmin, fminf(fminf(v.x, v.y), fminf(v.z, v.w)));
      lmax = fmaxf(lmax, fmaxf(fmaxf(v.x, v.y), fmaxf(v.z, v.w)));
    }
    float rs = wave_sum_wmma(ls);
    float rq = wave_sum_wmma(lsq);
    float rn = wave_min(lmin);
    float rx = wave_max(lmax);
    if (lane == 0) {
      st[t][wv][0] = rs; st[t][wv][1] = rq; st[t][wv][2] = rn; st[t][wv][3] = rx;
    }
  }

  if (GT_WS) {
#pragma unroll
    for (int i = 0; i < 4; ++i)
      gt4[(size_t)b * kHW4 + base4 + tid + i * 256] = gacc[i] * 0.0625f;
  }

  __syncthreads();
  if (tid < 64) {
    const int t = tid >> 2, q = tid & 3;
    float v = st[t][0][q];
#pragma unroll
    for (int w = 1; w < 8; ++w) {
      float u = st[t][w][q];
      v = (q < 2) ? (v + u) : (q == 2 ? fminf(v, u) : fmaxf(v, u));
    }
    const int idx = t * kB + b;
    if      (q == 0) atomicAdd(&sumS[idx], v);
    else if (q == 1) atomicAdd(&sqS[idx], v);
    else if (q == 2) atomicMin(&minS[idx], fkey(v));
    else             atomicMax(&maxS[idx], fkey(v));
  }
}

// ---------------------------------------------------------------------------
// K2: per-b stats of the gt map (14 MB, L2-resident).
// ---------------------------------------------------------------------------
template <bool GT_WS>
__global__ __launch_bounds__(256) void k_gt_stats(const float* __restrict__ xf,
                                                  float* __restrict__ ws) {
  const f4* x4   = (const f4*)xf;
  const f4* gt4  = (const f4*)(ws + OFF_GT);
  float*    sumG = ws + OFF_SUM_G;
  float*    sqG  = ws + OFF_SQ_G;
  unsigned* minG = (unsigned*)(ws + OFF_MIN_G);
  unsigned* maxG = (unsigned*)(ws + OFF_MAX_G);

  const int tid = threadIdx.x, lane = tid & 31, wv = tid >> 5;
  const int b = blockIdx.y, chunk = blockIdx.x;   // 9 chunks x 3072 f4
  const int base4 = chunk * 3072;

  float ls = 0.f, lsq = 0.f, lmin = 3.402823e38f, lmax = -3.402823e38f;
#pragma unroll 1
  for (int i = 0; i < 12; ++i) {
    int p4 = base4 + i * 256 + tid;
    f4 g = load_gt<GT_WS>(x4, gt4, b, p4);
    ls  += (g.x + g.y) + (g.z + g.w);
    lsq += (g.x * g.x + g.y * g.y) + (g.z * g.z + g.w * g.w);
    lmin = fminf(lmin, fminf(fminf(g.x, g.y), fminf(g.z, g.w)));
    lmax = fmaxf(lmax, fmaxf(fmaxf(g.x, g.y), fmaxf(g.z, g.w)));
  }
  float rs = wave_sum_wmma(ls);
  float rq = wave_sum_wmma(lsq);
  float rn = wave_min(lmin);
  float rx = wave_max(lmax);

  __shared__ float sh[8][4];
  if (lane == 0) { sh[wv][0] = rs; sh[wv][1] = rq; sh[wv][2] = rn; sh[wv][3] = rx; }
  __syncthreads();
  if (tid == 0) {
    float a = 0.f, s2 = 0.f, mn = 3.402823e38f, mx = -3.402823e38f;
#pragma unroll
    for (int w = 0; w < 8; ++w) {
      a += sh[w][0]; s2 += sh[w][1];
      mn = fminf(mn, sh[w][2]); mx = fmaxf(mx, sh[w][3]);
    }
    atomicAdd(&sumG[b], a);
    atomicAdd(&sqG[b], s2);
    atomicMin(&minG[b], fkey(mn));
    atomicMax(&maxG[b], fkey(mx));
  }
}

// ---------------------------------------------------------------------------
// K3: second streaming pass: per-(t,b) Sum(s*gt) and sim = Sum(min(dist)).
//     dist_m = (m - lo_m) / (Sum_m - N*lo_m)   [(hi-lo) cancels analytically]
// ---------------------------------------------------------------------------
template <bool GT_WS>
__global__ __launch_bounds__(256) void k_pass_c(const float* __restrict__ xf,
                                                float* __restrict__ ws) {
  const f4* x4  = (const f4*)xf;
  const f4* gt4 = (const f4*)(ws + OFF_GT);
  float* prodA  = ws + OFF_PROD;
  float* simA   = ws + OFF_SIM;

  const int tid = threadIdx.x, lane = tid & 31, wv = tid >> 5;
  const int chunk = blockIdx.x;        // 0..3
  const int t     = blockIdx.y;        // 0..15
  const int b     = blockIdx.z;        // 0..31
  const int idx   = t * kB + b;

  const float sum_s = ws[OFF_SUM_S + idx];
  const float lo_s  = funkey(((const unsigned*)(ws + OFF_MIN_S))[idx]);
  const float sum_g = ws[OFF_SUM_G + b];
  const float lo_g  = funkey(((const unsigned*)(ws + OFF_MIN_G))[b]);
  const float as = 1.0f / (sum_s - kN * lo_s);
  const float ag = 1.0f / (sum_g - kN * lo_g);

  const int base4 = chunk * 6912;
  float lp = 0.f, lsim = 0.f;
#pragma unroll 1
  for (int i = 0; i < 27; ++i) {
    int p4 = base4 + i * 256 + tid;
    f4 s = ntload(x4 + (size_t)(b * kT + t) * kHW4 + p4);
    f4 g = load_gt<GT_WS>(x4, gt4, b, p4);
    lp += (s.x * g.x + s.y * g.y) + (s.z * g.z + s.w * g.w);
    f4 ds = (s - lo_s) * as;
    f4 dg = (g - lo_g) * ag;
    lsim += (fminf(ds.x, dg.x) + fminf(ds.y, dg.y)) +
            (fminf(ds.z, dg.z) + fminf(ds.w, dg.w));
  }
  float rp = wave_sum_wmma(lp);
  float rs = wave_sum_wmma(lsim);

  __shared__ float sh[8][2];
  if (lane == 0) { sh[wv][0] = rp; sh[wv][1] = rs; }
  __syncthreads();
  if (tid == 0) {
    float a = 0.f, c = 0.f;
#pragma unroll
    for (int w = 0; w < 8; ++w) { a += sh[w][0]; c += sh[w][1]; }
    atomicAdd(&prodA[idx], a);
    atomicAdd(&simA[idx], c);
  }
}

// ---------------------------------------------------------------------------
// K4: 512 scalar finishes.  cc = cov / sqrt(SS_s * SS_g)   (ddof cancels)
// ---------------------------------------------------------------------------
__global__ __launch_bounds__(256) void k_final(const float* __restrict__ ws,
                                               float* __restrict__ out) {
  int i = blockIdx.x * 256 + threadIdx.x;
  if (i < 512) {
    const int bb = i & 31;
    float sum_s = ws[OFF_SUM_S + i];
    float sq_s  = ws[OFF_SQ_S + i];
    float sum_g = ws[OFF_SUM_G + bb];
    float sq_g  = ws[OFF_SQ_G + bb];
    float prod  = ws[OFF_PROD + i];
    float sim   = ws[OFF_SIM + i];

    float ms = sum_s / kN, mg = sum_g / kN;
    float ss  = sq_s - kN * ms * ms;     // Sum((s-ms)^2)
    float sg  = sq_g - kN * mg * mg;     // Sum((g-mg)^2)
    float cov = prod - kN * ms * mg;     // Sum((s-ms)(g-mg))

    out[i]       = cov / sqrtf(ss * sg); // cc [16,32] flat
    out[512 + i] = sim;                  // sim [16,32] flat
  }
}

// ---------------------------------------------------------------------------
extern "C" void kernel_launch(void* const* d_in, const int* in_sizes, int n_in,
                              void* d_out, int out_size, void* d_ws, size_t ws_size,
                              hipStream_t stream) {
  (void)in_sizes; (void)n_in; (void)out_size;
  const float* x  = (const float*)d_in[0];
  float*       out = (float*)d_out;
  float*       ws  = (float*)d_ws;

  const size_t need_full = (size_t)(OFF_GT + kB * kHW) * sizeof(float);
  const bool gt_in_ws = (ws_size >= need_full);

  k_init<<<(INIT_WORDS + 255) / 256, 256, 0, stream>>>((unsigned*)d_ws);

  if (gt_in_ws) {
    k_pass_a<true ><<<dim3(27, 32),     256, 0, stream>>>(x, ws);
    k_gt_stats<true ><<<dim3(9, 32),    256, 0, stream>>>(x, ws);
    k_pass_c<true ><<<dim3(4, 16, 32),  256, 0, stream>>>(x, ws);
  } else {
    k_pass_a<false><<<dim3(27, 32),     256, 0, stream>>>(x, ws);
    k_gt_stats<false><<<dim3(9, 32),    256, 0, stream>>>(x, ws);
    k_pass_c<false><<<dim3(4, 16, 32),  256, 0, stream>>>(x, ws);
  }
  k_final<<<2, 256, 0, stream>>>(ws, out);
}